// SimpleGraphSAGE_69088843924164
// MI455X (gfx1250) — compile-verified
//
#include <hip/hip_runtime.h>
#include <hip/hip_bf16.h>

typedef __attribute__((ext_vector_type(2))) float v2f;
typedef __attribute__((ext_vector_type(8))) float v8f;

#define D 128
#define LDS_STRIDE 132   // 132 % 64 == 4 -> banks (row*132 + k) distinct across 16 rows; 132*4 % 16 == 0 keeps float4/b64 alignment

// ---------------- Kernel 1: zero summed[N*128] + deg[N] ----------------
__global__ void sage_zero_ws(float* __restrict__ ws, long long total) {
    long long i = (long long)blockIdx.x * blockDim.x + threadIdx.x;
    long long stride = (long long)gridDim.x * blockDim.x;
    for (; i < total; i += stride) ws[i] = 0.0f;
}

// ---------------- Kernel 2: edge scatter (one wave32 per edge) ----------------
__global__ void sage_edge_scatter(const float* __restrict__ x,
                                  const int* __restrict__ ei,   // [2, E] int32
                                  float* __restrict__ summed,
                                  float* __restrict__ deg,
                                  int E) {
    int gwave = (int)((blockIdx.x * (unsigned)blockDim.x + threadIdx.x) >> 5);
    int lane  = threadIdx.x & 31;
    if (gwave >= E) return;                       // uniform per wave
    int src = ei[gwave];
    int dst = ei[(long long)E + gwave];
    const float4* xr = (const float4*)(x + (long long)src * D);
    float4 v = xr[lane];                          // lanes cover 128 floats, fully coalesced 512B
    float* sr = summed + (long long)dst * D + lane * 4;
    atomicAdd(sr + 0, v.x);
    atomicAdd(sr + 1, v.y);
    atomicAdd(sr + 2, v.z);
    atomicAdd(sr + 3, v.w);
    if (lane == 0) atomicAdd(&deg[dst], 1.0f);
}

// ---------------- Kernel 3: fused mean + dual WMMA GEMM + bias ----------------
// out[m, n] = sum_k (summed[m,k]/max(deg[m],1))*Wl[n,k] + bl[n] + sum_k x[m,k]*Wr[n,k]
// Block: 256 threads (8 waves). Block owns 16 rows; wave w owns n-tile [16w, 16w+16).
__global__ void __launch_bounds__(256)
sage_gemm(const float* __restrict__ summed, const float* __restrict__ deg,
          const float* __restrict__ x,
          const float* __restrict__ Wl, const float* __restrict__ bl,
          const float* __restrict__ Wr, float* __restrict__ out, int nNodes) {
    __shared__ float aggT[16 * LDS_STRIDE];
    __shared__ float xT[16 * LDS_STRIDE];

    const int m0  = blockIdx.x * 16;
    const int tid = threadIdx.x;

    // Cooperative stage of the 16x128 agg / x tiles (each thread: 8 floats = 2 float4).
    // Mean (divide-by-degree) is fused here: one rcp per thread, then multiplies.
    {
        int row = tid >> 4;            // 0..15
        int col = (tid & 15) * 8;      // 0..120
        int gr  = m0 + row;
        float4 a0 = {0, 0, 0, 0}, a1 = a0, b0 = a0, b1 = a0;
        if (gr < nNodes) {
            const float4* ga = (const float4*)(summed + (long long)gr * D + col);
            const float4* gx = (const float4*)(x      + (long long)gr * D + col);
            a0 = ga[0]; a1 = ga[1];
            b0 = gx[0]; b1 = gx[1];
            float rd = 1.0f / fmaxf(deg[gr], 1.0f);
            a0.x *= rd; a0.y *= rd; a0.z *= rd; a0.w *= rd;
            a1.x *= rd; a1.y *= rd; a1.z *= rd; a1.w *= rd;
        }
        float4* la = (float4*)&aggT[row * LDS_STRIDE + col];
        float4* lx = (float4*)&xT[row * LDS_STRIDE + col];
        la[0] = a0; la[1] = a1;
        lx[0] = b0; lx[1] = b1;
    }
    __syncthreads();

    const int wave = tid >> 5;             // 0..7 -> n-tile
    const int lane = tid & 31;
    const int n0   = wave * 16;
    const int lm   = lane & 15;            // M (for A) / N (for B) within tile
    const int kh   = (lane >> 4) * 2;      // K sub-offset: 0 (lanes 0-15) or 2 (lanes 16-31)

    // Accumulator pre-loaded with bias: C[r] holds M = r (+8 for hi lanes), N = lm
    float bias = bl[n0 + lm];
    v8f acc;
#pragma unroll
    for (int r = 0; r < 8; ++r) acc[r] = bias;

    const float* wlp = Wl + (long long)(n0 + lm) * D + kh;   // B lane holds column n = n0+lm
    const float* wrp = Wr + (long long)(n0 + lm) * D + kh;
    const float* lap = &aggT[lm * LDS_STRIDE + kh];          // A lane holds row m = lm
    const float* lxp = &xT[lm * LDS_STRIDE + kh];

#pragma unroll 8
    for (int k0 = 0; k0 < D; k0 += 4) {
        v2f aA = *(const v2f*)(lap + k0);
        v2f aX = *(const v2f*)(lxp + k0);
        v2f bL = *(const v2f*)(wlp + k0);
        v2f bR = *(const v2f*)(wrp + k0);
        acc = __builtin_amdgcn_wmma_f32_16x16x4_f32(false, aA, false, bL,
                                                    (short)0, acc, false, false);
        acc = __builtin_amdgcn_wmma_f32_16x16x4_f32(false, aX, false, bR,
                                                    (short)0, acc, false, false);
    }

    // Store: VGPR r -> row m0 + r + 8*(lane>=16), column n0 + lm
    const int mBase = m0 + ((lane >> 4) << 3);
    float* op = out + (long long)mBase * D + n0 + lm;
    if (m0 + 16 <= nNodes) {
        // Uniform fast path: full tile in range, no per-lane exec masking
#pragma unroll
        for (int r = 0; r < 8; ++r) op[(long long)r * D] = acc[r];
    } else {
#pragma unroll
        for (int r = 0; r < 8; ++r) {
            if (mBase + r < nNodes) op[(long long)r * D] = acc[r];
        }
    }
}

extern "C" void kernel_launch(void* const* d_in, const int* in_sizes, int n_in,
                              void* d_out, int out_size, void* d_ws, size_t ws_size,
                              hipStream_t stream) {
    const float* x  = (const float*)d_in[0];   // [N, 128]
    const int*   ei = (const int*)d_in[1];     // [2, E]
    const float* Wl = (const float*)d_in[2];   // [128, 128]
    const float* bl = (const float*)d_in[3];   // [128]
    const float* Wr = (const float*)d_in[4];   // [128, 128]
    float* out = (float*)d_out;

    const int N = in_sizes[0] / D;
    const int E = in_sizes[1] / 2;

    float* summed = (float*)d_ws;              // N*128 floats
    float* deg    = summed + (long long)N * D; // N floats
    const long long totWs = (long long)N * D + N;

    // 1. zero workspace (summed + deg must start at 0 for the atomic accumulate)
    {
        long long blocks = (totWs + 255) / 256;
        if (blocks > 65535) blocks = 65535;
        sage_zero_ws<<<(int)blocks, 256, 0, stream>>>((float*)d_ws, totWs);
    }
    // 2. edge scatter: one wave per edge, 8 waves per 256-thread block
    {
        int blocks = (E + 7) / 8;
        sage_edge_scatter<<<blocks, 256, 0, stream>>>(x, ei, summed, deg, E);
    }
    // 3. fused mean + WMMA GEMM + bias
    {
        int blocks = (N + 15) / 16;
        sage_gemm<<<blocks, 256, 0, stream>>>(summed, deg, x, Wl, bl, Wr, out, N);
    }
}